// SSM2D_66571993089002
// MI455X (gfx1250) — compile-verified
//
#include <hip/hip_runtime.h>
#include <hip/hip_bf16.h>

// Problem constants (from reference)
#define B_SZ  2
#define CIN   96
#define HH    128
#define WW    128
#define HW    16384          // HH*WW
#define NTOT  32768          // B_SZ*HW
#define DTR   8              // DT_RANK
#define NST   16             // D_STATE
#define DIN   96             // D_INNER
#define OPROJ 40             // DTR + 2*NST
#define OPAD  48             // OPROJ padded to 3x16 row tiles

typedef __attribute__((ext_vector_type(2))) float v2f;
typedef __attribute__((ext_vector_type(8))) float v8f;

// ---------------------------------------------------------------------------
// K0: transpose x (b,c,h,w) -> xT[p][c], p = b*HW + h*WW + w. LDS-tiled so both
// global accesses are coalesced. Working set stays L2-resident (12.6 MB).
// ---------------------------------------------------------------------------
__global__ __launch_bounds__(256) void k0_transpose(const float* __restrict__ x,
                                                    float* __restrict__ xT) {
    __shared__ float tile[32][33];
    const int tx = threadIdx.x;            // 0..31
    const int ty = threadIdx.y;            // 0..7
    const int p0 = blockIdx.x << 5;        // 32-pixel tile (never crosses batch: HW%32==0)
    const int c0 = blockIdx.y << 5;        // 0,32,64
    const int b  = p0 >> 14;
    const int pl = p0 & (HW - 1);
#pragma unroll
    for (int q = 0; q < 4; ++q) {
        int c = c0 + ty + q * 8;
        tile[ty + q * 8][tx] = x[((size_t)(b * CIN + c)) * HW + pl + tx];
    }
    __syncthreads();
#pragma unroll
    for (int q = 0; q < 4; ++q) {
        int pr = ty + q * 8;
        xT[(size_t)(p0 + pr) * CIN + c0 + tx] = tile[tx][pr];
    }
}

// ---------------------------------------------------------------------------
// KA: Aneg[d][n] = -exp(A_log[d][n])
// ---------------------------------------------------------------------------
__global__ void kA_prep(const float* __restrict__ A_log, float* __restrict__ Aneg) {
    int i = blockIdx.x * blockDim.x + threadIdx.x;
    if (i < DIN * NST) Aneg[i] = -__expf(A_log[i]);
}

// ---------------------------------------------------------------------------
// K1: dBC projection, fp32 WMMA 16x16x4.  One wave per 16-pixel tile.
//   D(40pad48 x 16pix) = Wdbc(40x96) * X(96x16pix) + bias
// Weights staged once per block into LDS, zero-padded to 48 rows, so the
// inner loop has only unconditional ds_load_b64 (A) + global_load_b32 (B)
// + 3x v_wmma — no EXEC divergence around the WMMAs.
// A layout (16x4 f32): lane L: M = L&15; half-wave picks K pair {k0,k0+1}/{k0+2,k0+3}.
// B layout (4x16 f32): lane L: N = L&15; same half-wave K-pair split.
// C/D layout: vgpr r: lanes0-15 -> (M=r, N=lane), lanes16-31 -> (M=8+r, N=lane-16).
// Rows 0..7 -> deltaRaw (channel-major [8][NTOT]); 8..23 -> Bm[p][16]; 24..39 -> Cm[p][16].
// ---------------------------------------------------------------------------
__global__ __launch_bounds__(256) void k1_dbc_proj(const float* __restrict__ x,
        const float* __restrict__ Wdbc, const float* __restrict__ bdbc,
        float* __restrict__ deltaRaw, float* __restrict__ Bm, float* __restrict__ Cm) {
    __shared__ float Wlds[OPAD * CIN];     // 48*96 floats = 18 KB, rows 40..47 zero
    for (int idx = threadIdx.x; idx < OPAD * CIN; idx += 256)
        Wlds[idx] = (idx < OPROJ * CIN) ? Wdbc[idx] : 0.0f;
    __syncthreads();

    const int lane = threadIdx.x & 31;
    const int wave = blockIdx.x * (blockDim.x >> 5) + (threadIdx.x >> 5);
    const int p0   = wave << 4;                  // 16 pixels per wave
    const int b    = p0 >> 14;
    const int pl   = p0 & (HW - 1);
    const int col  = lane & 15;
    const int koff = (lane < 16) ? 0 : 2;

    v8f acc0 = {}; v8f acc1 = {}; v8f acc2 = {};
    const float* xb = x + (size_t)(b * CIN) * HW + pl + col;
    const float* w0 = Wlds + (0  + col) * CIN + koff;
    const float* w1 = Wlds + (16 + col) * CIN + koff;
    const float* w2 = Wlds + (32 + col) * CIN + koff;

    for (int k0 = 0; k0 < CIN; k0 += 4) {
        v2f bv; bv.x = xb[(size_t)(k0 + koff) * HW]; bv.y = xb[(size_t)(k0 + koff + 1) * HW];
        v2f a0, a1, a2;
        a0.x = w0[k0]; a0.y = w0[k0 + 1];
        a1.x = w1[k0]; a1.y = w1[k0 + 1];
        a2.x = w2[k0]; a2.y = w2[k0 + 1];
        acc0 = __builtin_amdgcn_wmma_f32_16x16x4_f32(false, a0, false, bv, (short)0, acc0, false, false);
        acc1 = __builtin_amdgcn_wmma_f32_16x16x4_f32(false, a1, false, bv, (short)0, acc1, false, false);
        acc2 = __builtin_amdgcn_wmma_f32_16x16x4_f32(false, a2, false, bv, (short)0, acc2, false, false);
    }

    const int rb = (lane < 16) ? 0 : 8;
    const int p  = p0 + col;
    float vals[24];
#pragma unroll
    for (int r = 0; r < 8; ++r) { vals[r] = acc0[r]; vals[8 + r] = acc1[r]; vals[16 + r] = acc2[r]; }
#pragma unroll
    for (int g = 0; g < 3; ++g) {
#pragma unroll
        for (int r = 0; r < 8; ++r) {
            const int row = g * 16 + rb + r;
            if (row >= OPROJ) continue;            // compile-time prune (fully unrolled)
            const float v = vals[g * 8 + r] + bdbc[row];
            if (row < DTR)           deltaRaw[(size_t)row * NTOT + p]        = v;
            else if (row < DTR+NST)  Bm[(size_t)p * NST + (row - DTR)]       = v;
            else                     Cm[(size_t)p * NST + (row - DTR - NST)] = v;
        }
    }
}

// ---------------------------------------------------------------------------
// K2: delta = softplus(dt_w(96x8) * deltaRaw(8xN) + dt_b), fp32 WMMA 16x16x4,
// two K-steps (K=8), six 16-row tiles, dt_w staged in LDS.
// Output pixel-major delta[p][96].
// ---------------------------------------------------------------------------
__global__ __launch_bounds__(256) void k2_dt_proj(const float* __restrict__ deltaRaw,
        const float* __restrict__ dtw, const float* __restrict__ dtb,
        float* __restrict__ delta) {
    __shared__ float Wlds[DIN * DTR];      // 768 floats
    for (int idx = threadIdx.x; idx < DIN * DTR; idx += 256)
        Wlds[idx] = dtw[idx];
    __syncthreads();

    const int lane = threadIdx.x & 31;
    const int wave = blockIdx.x * (blockDim.x >> 5) + (threadIdx.x >> 5);
    const int p0   = wave << 4;
    const int col  = lane & 15;
    const int koff = (lane < 16) ? 0 : 2;
    const int rb   = (lane < 16) ? 0 : 8;
    const int p    = p0 + col;

    v2f bv0, bv1;
    bv0.x = deltaRaw[(size_t)(0 + koff) * NTOT + p];
    bv0.y = deltaRaw[(size_t)(1 + koff) * NTOT + p];
    bv1.x = deltaRaw[(size_t)(4 + koff) * NTOT + p];
    bv1.y = deltaRaw[(size_t)(5 + koff) * NTOT + p];

    for (int d0 = 0; d0 < DIN; d0 += 16) {
        const float* wrow = Wlds + (d0 + col) * DTR + koff;
        v2f a0, a1;
        a0.x = wrow[0]; a0.y = wrow[1];
        a1.x = wrow[4]; a1.y = wrow[5];
        v8f acc = {};
        acc = __builtin_amdgcn_wmma_f32_16x16x4_f32(false, a0, false, bv0, (short)0, acc, false, false);
        acc = __builtin_amdgcn_wmma_f32_16x16x4_f32(false, a1, false, bv1, (short)0, acc, false, false);
#pragma unroll
        for (int r = 0; r < 8; ++r) {
            const int row = d0 + rb + r;
            const float z = acc[r] + dtb[row];
            const float sp = fmaxf(z, 0.0f) + log1pf(__expf(-fabsf(z)));   // stable softplus
            delta[(size_t)p * DIN + row] = sp;
        }
    }
}

// ---------------------------------------------------------------------------
// K3: closed-form diagonal scan + output contraction.
// hs[i,j,d,n] = sum_{t<L} exp(Aneg[d,n]*S_t) * delta[p-129t,d]*Bm[p-129t,n]*xT[p-129t,d]
// with L = 2^min(ctz(i+1),ctz(j+1)); y = sum_n hs*Cm[p,n] + D[d]*xT[p,d].
// One 96-thread block per pixel (thread = d). Avoids materializing the
// reference's 2x201MB A_bar/BX tensors entirely.
// ---------------------------------------------------------------------------
__global__ __launch_bounds__(96) void k3_scan_out(const float* __restrict__ xT,
        const float* __restrict__ delta, const float* __restrict__ Bm,
        const float* __restrict__ Cm, const float* __restrict__ Aneg,
        const float* __restrict__ Dvec, float* __restrict__ out) {
    const int p = blockIdx.x;
    const int d = threadIdx.x;
    const int pl = p & (HW - 1);
    const int i = pl >> 7, j = pl & (WW - 1);
    const int m = min(__ffs(i + 1) - 1, __ffs(j + 1) - 1);   // ctz
    const int L = 1 << m;

    float Arow[NST];
#pragma unroll
    for (int n = 0; n < NST; ++n) Arow[n] = Aneg[d * NST + n];

    float h[NST];
#pragma unroll
    for (int n = 0; n < NST; ++n) h[n] = 0.0f;

    float S = 0.0f;
    for (int t = 0; t < L; ++t) {
        const int pt = p - t * (WW + 1);              // diagonal predecessor
        const float dl = delta[(size_t)pt * DIN + d]; // coalesced over d
        const float g  = dl * xT[(size_t)pt * DIN + d];
        const float* Bp = Bm + (size_t)pt * NST;      // broadcast across block
        if (t == 0) {
#pragma unroll
            for (int n = 0; n < NST; ++n) h[n] += g * Bp[n];
        } else {
#pragma unroll
            for (int n = 0; n < NST; ++n) h[n] += __expf(Arow[n] * S) * g * Bp[n];
        }
        S += dl;
    }

    const float* Cp = Cm + (size_t)p * NST;
    float y = 0.0f;
#pragma unroll
    for (int n = 0; n < NST; ++n) y += h[n] * Cp[n];
    y += Dvec[d] * xT[(size_t)p * DIN + d];
    out[(size_t)p * DIN + d] = y;
}

// ---------------------------------------------------------------------------
extern "C" void kernel_launch(void* const* d_in, const int* in_sizes, int n_in,
                              void* d_out, int out_size, void* d_ws, size_t ws_size,
                              hipStream_t stream) {
    const float* x     = (const float*)d_in[0];
    const float* Wdbc  = (const float*)d_in[1];
    const float* bdbc  = (const float*)d_in[2];
    const float* dtw   = (const float*)d_in[3];
    const float* dtb   = (const float*)d_in[4];
    const float* Alog  = (const float*)d_in[5];
    const float* Dv    = (const float*)d_in[6];
    float* out = (float*)d_out;

    // Workspace layout (floats): total ~7.6M floats = 30.4 MB
    float* ws       = (float*)d_ws;
    float* xT       = ws;                               // NTOT*96
    float* deltaRaw = xT + (size_t)NTOT * CIN;          // 8*NTOT
    float* Bm       = deltaRaw + (size_t)DTR * NTOT;    // NTOT*16
    float* Cm       = Bm + (size_t)NTOT * NST;          // NTOT*16
    float* delta    = Cm + (size_t)NTOT * NST;          // NTOT*96
    float* Aneg     = delta + (size_t)NTOT * DIN;       // 96*16

    k0_transpose<<<dim3(NTOT / 32, CIN / 32), dim3(32, 8), 0, stream>>>(x, xT);
    kA_prep<<<(DIN * NST + 255) / 256, 256, 0, stream>>>(Alog, Aneg);
    k1_dbc_proj<<<NTOT / 16 / 8, 256, 0, stream>>>(x, Wdbc, bdbc, deltaRaw, Bm, Cm);
    k2_dt_proj<<<NTOT / 16 / 8, 256, 0, stream>>>(deltaRaw, dtw, dtb, delta);
    k3_scan_out<<<NTOT, 96, 0, stream>>>(xT, delta, Bm, Cm, Aneg, Dv, out);
}